// OHEMLoss_28054726378143
// MI455X (gfx1250) — compile-verified
//
#include <hip/hip_runtime.h>
#include <hip/hip_bf16.h>

typedef float v2f __attribute__((ext_vector_type(2)));
typedef float v8f __attribute__((ext_vector_type(8)));

#define IGNORE_INDEX (-100)
#define ORD_NEGINF 0x007FFFFFu   // ordf(-inf)
#define RBLOCKS 1024

struct Ctl {
    int      pos;       // count of label != 0
    unsigned prefix;    // radix-select prefix (high bytes fixed)
    unsigned k;         // remaining rank within current prefix
    unsigned thr_u;     // final threshold (ordered-uint space)
    int      branch;    // 1 if num_neg > neg_sum (threshold branch taken)
    int      pad[3];
    unsigned hist[4][256];
};

// Order-preserving float -> uint map (ascending float == ascending uint)
__device__ __forceinline__ unsigned ordf(float f) {
    unsigned b = __float_as_uint(f);
    return (b & 0x80000000u) ? ~b : (b | 0x80000000u);
}

__global__ __launch_bounds__(1024) void ohem_init(Ctl* ctl) {
    int t = threadIdx.x;
    if (t == 0) {
        ctl->pos = 0; ctl->prefix = 0; ctl->k = 1; ctl->thr_u = 0; ctl->branch = 0;
    }
    if (t < 1024) ctl->hist[t >> 8][t & 255] = 0;
}

// Pass 1: stream pred once. Per 16-row tile (one wave32):
//   - lanes (L, L+16) split row (L&15): cols {4s,4s+1} vs {4s+2,4s+3}, s=0..7
//   - row max via lane-local max + shfl_xor(16)
//   - Sum exp(x - max) via 8x V_WMMA_F32_16X16X4_F32 against a ones B-matrix
//   - nll[row] = max + log(sumexp) - pred[row][label]
//   - neg_u[row] = is_pos ? ord(-inf) : ord(max cols1..31)   (radix-select key)
//   - fused byte-3 histogram of neg_u into ctl->hist[3] via LDS
__global__ __launch_bounds__(256) void ohem_pass1(const float* __restrict__ pred,
                                                  const int* __restrict__ label,
                                                  unsigned* __restrict__ neg_u,
                                                  float* __restrict__ nll,
                                                  Ctl* __restrict__ ctl, int n) {
    __shared__ unsigned hh[256];
    hh[threadIdx.x] = 0;
    __syncthreads();

    const int lane   = threadIdx.x & 31;
    const int wib    = threadIdx.x >> 5;
    const int wpb    = blockDim.x >> 5;
    const int gwave  = blockIdx.x * wpb + wib;
    const int nwaves = gridDim.x * wpb;
    const int half   = lane >> 4;      // 0: K-offsets {0,1}, 1: {2,3}
    const int r      = lane & 15;      // row within tile for loading
    const int ntiles = n >> 4;

    for (int tile = gwave; tile < ntiles; tile += nwaves) {
        const int row = (tile << 4) + r;
        const float* rp = pred + (size_t)row * 32 + (half << 1);

        v2f v[8];
        #pragma unroll
        for (int s = 0; s < 8; ++s) v[s] = *(const v2f*)(rp + 4 * s);

        // max over all 32 columns (for softmax stability)
        float mloc = fmaxf(v[0].x, v[0].y);
        #pragma unroll
        for (int s = 1; s < 8; ++s) mloc = fmaxf(mloc, fmaxf(v[s].x, v[s].y));
        float mall = fmaxf(mloc, __shfl_xor(mloc, 16, 32));

        // max over columns 1..31 (col 0 is v[0].x of half==0)
        float mnb_loc = (half == 0) ? v[0].y : fmaxf(v[0].x, v[0].y);
        #pragma unroll
        for (int s = 1; s < 8; ++s) mnb_loc = fmaxf(mnb_loc, fmaxf(v[s].x, v[s].y));
        float mnb = fmaxf(mnb_loc, __shfl_xor(mnb_loc, 16, 32));

        // exp(x - max), still in the WMMA A layout
        v2f e[8];
        #pragma unroll
        for (int s = 0; s < 8; ++s) {
            e[s].x = expf(v[s].x - mall);
            e[s].y = expf(v[s].y - mall);
        }

        // Row sums via WMMA: D = sum_K A * ones.  A: 16x4 f32, B: 4x16 ones.
        v8f acc = {0.f, 0.f, 0.f, 0.f, 0.f, 0.f, 0.f, 0.f};
        v2f ones = {1.0f, 1.0f};
        #pragma unroll
        for (int s = 0; s < 8; ++s) {
            acc = __builtin_amdgcn_wmma_f32_16x16x4_f32(
                false, e[s], false, ones, (short)0, acc, false, false);
        }

        // D layout: acc[j] holds rowsum(j) on lanes 0..15 and rowsum(j+8) on 16..31.
        // Writers: lanes 0..7 -> rows 0..7 (acc[lane]); lanes 16..23 -> rows 8..15 (acc[lane-16]).
        const int  j   = lane & 7;
        const bool isw = (lane < 8) || (lane >= 16 && lane < 24);
        float se = acc[0];
        se = (j == 1) ? acc[1] : se;
        se = (j == 2) ? acc[2] : se;
        se = (j == 3) ? acc[3] : se;
        se = (j == 4) ? acc[4] : se;
        se = (j == 5) ? acc[5] : se;
        se = (j == 6) ? acc[6] : se;
        se = (j == 7) ? acc[7] : se;

        // maxes live on lane (row&15); writer lane w needs row (w<16 ? w : w-8)
        const int src   = (lane < 16) ? lane : (lane - 8);
        float mall_w = __shfl(mall, src, 32);
        float mnb_w  = __shfl(mnb,  src, 32);

        bool posflag = false;
        if (isw) {
            const int rw  = (tile << 4) + ((lane < 16) ? lane : (lane - 8));
            const int lab = label[rw];
            const int labc = min(max(lab, 0), 31);
            const float plab = pred[(size_t)rw * 32 + labc];
            nll[rw] = (mall_w + logf(se)) - plab;
            posflag = (lab != 0);
            const unsigned u = posflag ? ORD_NEGINF : ordf(mnb_w);
            neg_u[rw] = u;
            atomicAdd(&hh[u >> 24], 1u);   // fused byte-3 radix histogram
        }
        int cnt = (int)__popcll(__ballot(posflag));
        if (lane == 0 && cnt) atomicAdd(&ctl->pos, cnt);
    }

    // scalar tail for n % 16 != 0 (no-op for N = 2^21)
    for (int row = (ntiles << 4) + gwave * 32 + lane; row < n; row += nwaves * 32) {
        const float* rp = pred + (size_t)row * 32;
        float m = rp[0];
        for (int c = 1; c < 32; ++c) m = fmaxf(m, rp[c]);
        float mnb = rp[1];
        for (int c = 2; c < 32; ++c) mnb = fmaxf(mnb, rp[c]);
        float se = 0.f;
        for (int c = 0; c < 32; ++c) se += expf(rp[c] - m);
        const int lab = label[row];
        const int labc = min(max(lab, 0), 31);
        nll[row] = (m + logf(se)) - rp[labc];
        const unsigned u = (lab != 0) ? ORD_NEGINF : ordf(mnb);
        neg_u[row] = u;
        atomicAdd(&hh[u >> 24], 1u);
        if (lab != 0) atomicAdd(&ctl->pos, 1);
    }

    __syncthreads();
    if (hh[threadIdx.x]) atomicAdd(&ctl->hist[3][threadIdx.x], hh[threadIdx.x]);
}

// One radix round (bytes 2..0): 256-bin LDS histogram among entries matching prefix.
__global__ __launch_bounds__(256) void ohem_hist(const unsigned* __restrict__ neg_u,
                                                 Ctl* __restrict__ ctl, int n, int B) {
    __shared__ unsigned hh[256];
    hh[threadIdx.x] = 0;
    __syncthreads();
    const unsigned prefix = ctl->prefix;
    const unsigned hm = 0xFFFFFFFFu << ((B + 1) * 8);   // B <= 2 here
    const int stride = gridDim.x * blockDim.x;
    for (int i = blockIdx.x * blockDim.x + threadIdx.x; i < n; i += stride) {
        const unsigned u = neg_u[i];
        if ((u & hm) == (prefix & hm))
            atomicAdd(&hh[(u >> (B * 8)) & 255u], 1u);
    }
    __syncthreads();
    if (hh[threadIdx.x]) atomicAdd(&ctl->hist[B][threadIdx.x], hh[threadIdx.x]);
}

// Suffix-scan the 256 bins from the top; pick bin containing the k-th largest.
__global__ __launch_bounds__(256) void ohem_scan(Ctl* __restrict__ ctl,
                                                 const int* __restrict__ factorp,
                                                 int n, int B) {
    __shared__ unsigned s[256];
    __shared__ unsigned h0[256];
    const int t = threadIdx.x;
    if (B == 3 && t == 0) {
        const long long pos     = ctl->pos;
        const long long neg_sum = pos * (long long)(*factorp);
        const long long num_neg = (long long)n - pos;
        ctl->branch = (num_neg > neg_sum) ? 1 : 0;
        long long k = neg_sum;
        if (k < 1) k = 1;
        if (k > n) k = n;
        ctl->k = (unsigned)k;
    }
    __syncthreads();
    const unsigned k = ctl->k;
    unsigned h = ctl->hist[B][t];
    h0[t] = h;
    s[t]  = h;
    __syncthreads();
    for (int off = 1; off < 256; off <<= 1) {
        unsigned add = (t + off < 256) ? s[t + off] : 0u;
        __syncthreads();
        s[t] += add;
        __syncthreads();
    }
    const unsigned incl = s[t];          // count of entries in bins >= t
    const unsigned excl = incl - h0[t];  // count in bins > t
    if (incl >= k && excl < k) {         // unique winner (k >= 1, total >= k)
        const unsigned np = ctl->prefix | ((unsigned)t << (B * 8));
        ctl->prefix = np;
        ctl->k = k - excl;
        if (B == 0) ctl->thr_u = np;
    }
}

// Deterministic masked reduction: fixed-order per-thread + fixed LDS tree.
// Note: for positive rows neg_u == ord(-inf) but the mask OR's in is_pos, so
// using the substituted key is exactly equivalent to comparing raw pred_value.
__global__ __launch_bounds__(256) void ohem_reduce(const unsigned* __restrict__ neg_u,
                                                   const float* __restrict__ nll,
                                                   const int* __restrict__ label,
                                                   const Ctl* __restrict__ ctl,
                                                   float* __restrict__ psum,
                                                   int* __restrict__ pcnt, int n) {
    __shared__ float ss[256];
    __shared__ int   sc[256];
    const int      branch = ctl->branch;
    const unsigned thr    = ctl->thr_u;
    float sum = 0.f;
    int   cnt = 0;
    const int stride = gridDim.x * blockDim.x;
    for (int i = blockIdx.x * blockDim.x + threadIdx.x; i < n; i += stride) {
        const int lab = label[i];
        const bool mask = branch ? ((neg_u[i] >= thr) || (lab != 0)) : (lab != -1);
        const bool valid = mask && (lab != IGNORE_INDEX);
        if (valid) { sum += nll[i]; ++cnt; }
    }
    const int t = threadIdx.x;
    ss[t] = sum; sc[t] = cnt;
    __syncthreads();
    for (int o = 128; o > 0; o >>= 1) {
        if (t < o) { ss[t] += ss[t + o]; sc[t] += sc[t + o]; }
        __syncthreads();
    }
    if (t == 0) { psum[blockIdx.x] = ss[0]; pcnt[blockIdx.x] = sc[0]; }
}

__global__ __launch_bounds__(256) void ohem_final(const float* __restrict__ psum,
                                                  const int* __restrict__ pcnt,
                                                  float* __restrict__ out, int nparts) {
    __shared__ float ss[256];
    __shared__ int   sc[256];
    const int t = threadIdx.x;
    float s = 0.f; int c = 0;
    for (int i = t; i < nparts; i += 256) { s += psum[i]; c += pcnt[i]; }
    ss[t] = s; sc[t] = c;
    __syncthreads();
    for (int o = 128; o > 0; o >>= 1) {
        if (t < o) { ss[t] += ss[t + o]; sc[t] += sc[t + o]; }
        __syncthreads();
    }
    if (t == 0) out[0] = ss[0] / (float)max(sc[0], 1);
}

extern "C" void kernel_launch(void* const* d_in, const int* in_sizes, int n_in,
                              void* d_out, int out_size, void* d_ws, size_t ws_size,
                              hipStream_t stream) {
    const float* pred   = (const float*)d_in[0];
    const int*   label  = (const int*)d_in[1];
    const int*   factor = (const int*)d_in[2];
    const int n = in_sizes[1];

    unsigned* neg_u = (unsigned*)d_ws;
    float*    nll   = (float*)(neg_u + n);
    Ctl*      ctl   = (Ctl*)(nll + n);
    float*    psum  = (float*)((char*)ctl + sizeof(Ctl));
    int*      pcnt  = (int*)(psum + RBLOCKS);

    ohem_init<<<1, 1024, 0, stream>>>(ctl);
    ohem_pass1<<<1024, 256, 0, stream>>>(pred, label, neg_u, nll, ctl, n);
    ohem_scan<<<1, 256, 0, stream>>>(ctl, factor, n, 3);   // hist[3] fused into pass1
    for (int B = 2; B >= 0; --B) {
        ohem_hist<<<512, 256, 0, stream>>>(neg_u, ctl, n, B);
        ohem_scan<<<1, 256, 0, stream>>>(ctl, factor, n, B);
    }
    ohem_reduce<<<RBLOCKS, 256, 0, stream>>>(neg_u, nll, label, ctl, psum, pcnt, n);
    ohem_final<<<1, 256, 0, stream>>>(psum, pcnt, (float*)d_out, RBLOCKS);
}